// PlainDecoder_41394894799455
// MI455X (gfx1250) — compile-verified
//
#include <hip/hip_runtime.h>
#include <hip/hip_bf16.h>
#include <math.h>
#include <stdint.h>

// ---------------------------------------------------------------------------
// Sizes from the reference
// ---------------------------------------------------------------------------
#define FEATURE_DIM 1024
#define EMBED_DIM   300
#define VOCAB       32000
#define HIDDEN      512
#define GATES4      (4 * HIDDEN)          // 2048
#define BATCH       64
#define MAX_LEN     40
#define TSTEPS      (MAX_LEN - 1)         // 39
#define XDIM        (EMBED_DIM + FEATURE_DIM) // 1324 (divisible by 4)

typedef __attribute__((ext_vector_type(2))) float v2f;
typedef __attribute__((ext_vector_type(8))) float v8f;
typedef unsigned int u32x4 __attribute__((ext_vector_type(4)));
typedef int          i32x4 __attribute__((ext_vector_type(4)));
typedef int          i32x8 __attribute__((ext_vector_type(8)));

#if defined(__has_builtin)
#if __has_builtin(__builtin_amdgcn_tensor_load_to_lds) && __has_builtin(__builtin_amdgcn_s_wait_tensorcnt)
#define HAVE_TDM 1
#endif
#endif
#ifndef HAVE_TDM
#define HAVE_TDM 0
#endif

// ---------------------------------------------------------------------------
// Workspace layout (element offsets)
// ---------------------------------------------------------------------------
#define WS_LT     0
#define WS_H      (WS_LT + BATCH * FEATURE_DIM)
#define WS_C      (WS_H  + BATCH * HIDDEN)
#define WS_GATES  (WS_C  + BATCH * HIDDEN)
#define WS_X      (WS_GATES + BATCH * GATES4)
#define WS_FLOATS (WS_X  + BATCH * XDIM)
#define WSI_SORT  0
#define WSI_DLEN  (WSI_SORT + BATCH)
#define WSI_CAPS  (WSI_DLEN + BATCH)

// Output layout: predictions [B][T][V] floats, then caps (B*MAX_LEN),
// then decode_lengths (B), then sort_ind (B), all as float.
#define OUT_PRED_ELEMS ((size_t)BATCH * TSTEPS * VOCAB)

// ---------------------------------------------------------------------------
// TDM: stage a 16-row x krows fp32 tile (row stride = krows floats, contiguous
// in LDS) from global memory into LDS via the Tensor Data Mover.
// Issued by the calling wave; waits TENSORcnt==0 before returning.
// D# layout per CDNA5 ISA ch.8.  6-arg builtin on this toolchain:
//   (u32x4 g0, i32x8 g1, i32x4 g2, i32x4 g3, i32x8 g4, i32 cpol)
// ---------------------------------------------------------------------------
__device__ __forceinline__ void tdm_load_tile_16rows(const float* gbase,
                                                     uint32_t lds_byte_addr,
                                                     uint32_t krows)
{
#if HAVE_TDM
    const uint64_t ga = (uint64_t)(uintptr_t)gbase;
    u32x4 g0;
    g0[0] = 1u;                                        // count=1 (valid user D#)
    g0[1] = lds_byte_addr;                             // lds_addr
    g0[2] = (uint32_t)(ga & 0xFFFFFFFFu);              // global_addr[31:0]
    g0[3] = (uint32_t)((ga >> 32) & 0x1FFFFFFu)        // global_addr[56:32]
          | (2u << 30);                                // type=2 ("image")
    const uint32_t td0 = krows;                        // tensor_dim0
    const uint32_t td1 = 16u;                          // tensor_dim1 (rows)
    const uint32_t s0  = krows;                        // tensor_dim0_stride
    i32x8 g1;
    g1[0] = (int)(2u << 16);                           // data_size=2 -> 4 bytes
    g1[1] = (int)((td0 & 0xFFFFu) << 16);              // tensor_dim0[15:0]
    g1[2] = (int)(((td0 >> 16) & 0xFFFFu)              // tensor_dim0[31:16]
          | ((td1 & 0xFFFFu) << 16));                  // tensor_dim1[15:0]
    g1[3] = (int)(((td1 >> 16) & 0xFFFFu)              // tensor_dim1[31:16]
          | ((krows & 0xFFFFu) << 16));                // tile_dim0
    g1[4] = (int)(16u);                                // tile_dim1=16, tile_dim2=0
    g1[5] = (int)(s0);                                 // tensor_dim0_stride[31:0]
    g1[6] = 0;                                         // stride0 hi | stride1 lo
    g1[7] = 0;                                         // stride1 hi
    i32x4 g2 = {0, 0, 0, 0};
    i32x4 g3 = {0, 0, 0, 0};
    i32x8 g4 = {0, 0, 0, 0, 0, 0, 0, 0};
    __builtin_amdgcn_tensor_load_to_lds(g0, g1, g2, g3, g4, 0);
    __builtin_amdgcn_s_wait_tensorcnt(0);
#else
    (void)gbase; (void)lds_byte_addr; (void)krows;
#endif
}

// Tell the compiler that `p` (an LDS buffer filled by the TDM behind its
// back) has been written: the pointer escapes into a volatile asm with a
// memory clobber, so subsequent LDS loads cannot be folded to undef.
__device__ __forceinline__ void lds_written_fence(const void* p)
{
    asm volatile("" : : "v"(p) : "memory");
}

// ---------------------------------------------------------------------------
// Kernel 1: stable argsort(-lengths) + gather caps + tail outputs
// ---------------------------------------------------------------------------
__global__ void prep_sort_kernel(const int* __restrict__ enc_caps,   // [B][MAX_LEN]
                                 const int* __restrict__ cap_len,    // [B][1]
                                 int* __restrict__ sort_ind,
                                 int* __restrict__ dec_len,
                                 int* __restrict__ caps_sorted,
                                 float* __restrict__ out_tail)
{
    __shared__ int len_sh[BATCH];
    const int b = threadIdx.x;
    len_sh[b] = cap_len[b];
    __syncthreads();
    const int myl = len_sh[b];
    int pos = 0;
    #pragma unroll 8
    for (int j = 0; j < BATCH; ++j) {
        const int lj = len_sh[j];
        if (lj > myl || (lj == myl && j < b)) ++pos;   // stable descending
    }
    sort_ind[pos] = b;
    __syncthreads();
    const int src = sort_ind[b];
    const int dl = len_sh[src] - 1;
    dec_len[b] = dl;
    for (int t = 0; t < MAX_LEN; ++t) {
        const int cv = enc_caps[src * MAX_LEN + t];
        caps_sorted[b * MAX_LEN + t] = cv;
        out_tail[b * MAX_LEN + t] = (float)cv;
    }
    out_tail[BATCH * MAX_LEN + b] = (float)dl;
    out_tail[BATCH * MAX_LEN + BATCH + b] = (float)src;
}

// ---------------------------------------------------------------------------
// Kernel 2: lt_sorted[i][f] = sum_s l_total[s][sort_ind[i]][f]; zero h, c
// ---------------------------------------------------------------------------
__global__ void prep_lt_kernel(const float* __restrict__ l_total,
                               const int* __restrict__ sort_ind,
                               float* __restrict__ lt_sorted,
                               float* __restrict__ h,
                               float* __restrict__ c)
{
    const int i = blockIdx.x;
    const int src = sort_ind[i];
    for (int f = threadIdx.x; f < FEATURE_DIM; f += blockDim.x) {
        const float s = l_total[(size_t)0 * BATCH * FEATURE_DIM + (size_t)src * FEATURE_DIM + f]
                      + l_total[(size_t)1 * BATCH * FEATURE_DIM + (size_t)src * FEATURE_DIM + f]
                      + l_total[(size_t)2 * BATCH * FEATURE_DIM + (size_t)src * FEATURE_DIM + f];
        lt_sorted[i * FEATURE_DIM + f] = s;
    }
    for (int j = threadIdx.x; j < HIDDEN; j += blockDim.x) {
        h[i * HIDDEN + j] = 0.0f;
        c[i * HIDDEN + j] = 0.0f;
    }
}

// ---------------------------------------------------------------------------
// Kernel 3: build x_t = concat(embed(caps[:, t]), lt_sorted)   [B][XDIM]
// ---------------------------------------------------------------------------
__global__ void build_x_kernel(const float* __restrict__ emb_weight,
                               const int* __restrict__ caps_sorted,
                               const float* __restrict__ lt_sorted,
                               int t,
                               float* __restrict__ x)
{
    const int i = blockIdx.x;
    const int tok = caps_sorted[i * MAX_LEN + t];
    for (int k = threadIdx.x; k < XDIM; k += blockDim.x) {
        const float v = (k < EMBED_DIM)
            ? emb_weight[(size_t)tok * EMBED_DIM + k]
            : lt_sorted[i * FEATURE_DIM + (k - EMBED_DIM)];
        x[i * XDIM + k] = v;
    }
}

// ---------------------------------------------------------------------------
// Kernel 4: gates = x @ W_ih^T + h @ W_hh^T + b_ih + b_hh  via fp32 WMMA.
// The h A-tile (16x512 fp32, shared by all 8 waves) is staged into LDS via
// the Tensor Data Mover; x and weight fragments stream from global/L2.
// Grid: (4 b-tiles, 16) x 256 threads (8 waves => 8 gate-tiles per block).
// ---------------------------------------------------------------------------
__global__ void gates_wmma_kernel(const float* __restrict__ x,    // [B][XDIM]
                                  const float* __restrict__ h,    // [B][HIDDEN]
                                  const float* __restrict__ W_ih, // [GATES4][XDIM]
                                  const float* __restrict__ W_hh, // [GATES4][HIDDEN]
                                  const float* __restrict__ b_ih,
                                  const float* __restrict__ b_hh,
                                  float* __restrict__ gates)      // [B][GATES4]
{
    __shared__ float htile[16 * HIDDEN];   // 32 KB

    const int lane  = threadIdx.x & 31;
    const int wave  = threadIdx.x >> 5;
    const int btile = blockIdx.x;                 // 0..3
    const int gtile = blockIdx.y * 8 + wave;      // 0..127
    const int half  = lane >> 4;                  // 0/1 -> K sub-pair
    const int l15   = lane & 15;
    const int arow  = btile * 16 + l15;           // A row (batch)
    const int bcol  = gtile * 16 + l15;           // B col (gate index)

#if HAVE_TDM
    if (threadIdx.x == 0) {
        tdm_load_tile_16rows(h + (size_t)btile * 16 * HIDDEN,
                             (uint32_t)(uintptr_t)htile, HIDDEN);
    }
#else
    for (int i = threadIdx.x; i < 16 * HIDDEN; i += blockDim.x)
        htile[i] = h[(size_t)btile * 16 * HIDDEN + i];
#endif
    __syncthreads();
    lds_written_fence(htile);   // TDM wrote LDS; keep the ds loads alive

    v8f acc = {};

    const float* xrow = x    + (size_t)arow * XDIM;
    const float* wrow = W_ih + (size_t)bcol * XDIM;
    #pragma unroll 4
    for (int k = 0; k < XDIM; k += 4) {
        const int ka = k + half * 2;
        if ((k & 127) == 0) {
            __builtin_prefetch(wrow + k + 256, 0, 1);
        }
        v2f a = *(const v2f*)(xrow + ka);
        v2f b = *(const v2f*)(wrow + ka);
        acc = __builtin_amdgcn_wmma_f32_16x16x4_f32(false, a, false, b,
                                                    (short)0, acc, false, false);
    }

    const float* hlds = htile + l15 * HIDDEN;
    const float* vrow = W_hh  + (size_t)bcol * HIDDEN;
    #pragma unroll 4
    for (int k = 0; k < HIDDEN; k += 4) {
        const int ka = k + half * 2;
        v2f a = *(const v2f*)(hlds + ka);   // ds_load_b64 from staged tile
        v2f b = *(const v2f*)(vrow + ka);
        acc = __builtin_amdgcn_wmma_f32_16x16x4_f32(false, a, false, b,
                                                    (short)0, acc, false, false);
    }

    const float bias = b_ih[bcol] + b_hh[bcol];
    #pragma unroll
    for (int r = 0; r < 8; ++r) {
        const int m = btile * 16 + r + half * 8;
        gates[(size_t)m * GATES4 + bcol] = acc[r] + bias;
    }
}

// ---------------------------------------------------------------------------
// Kernel 5: LSTM pointwise update (i,f,g,o split along gate dim)
// ---------------------------------------------------------------------------
__global__ void lstm_update_kernel(const float* __restrict__ gates,
                                   float* __restrict__ h,
                                   float* __restrict__ c)
{
    const int idx = blockIdx.x * blockDim.x + threadIdx.x;
    if (idx >= BATCH * HIDDEN) return;
    const int b = idx >> 9;
    const int j = idx & (HIDDEN - 1);
    const float* g = gates + (size_t)b * GATES4;
    const float gi = g[j];
    const float gf = g[HIDDEN + j];
    const float gg = g[2 * HIDDEN + j];
    const float go = g[3 * HIDDEN + j];
    const float i_ = 1.0f / (1.0f + __expf(-gi));
    const float f_ = 1.0f / (1.0f + __expf(-gf));
    const float g_ = tanhf(gg);
    const float o_ = 1.0f / (1.0f + __expf(-go));
    const float cc = f_ * c[idx] + i_ * g_;
    c[idx] = cc;
    h[idx] = o_ * tanhf(cc);
}

// ---------------------------------------------------------------------------
// Kernel 6: pred_t = h @ wdc_W^T + wdc_b, masked, stored into predictions.
// h A-tile staged into LDS via TDM (reused by 8 waves); wdc_W streams from
// L2 (65.5 MB, resident after step 0).
// Grid: (4 b-tiles, 250) x 256 threads (8 waves => 8 vocab-tiles per block).
// ---------------------------------------------------------------------------
__global__ void pred_wmma_kernel(const float* __restrict__ h,     // [B][HIDDEN]
                                 const float* __restrict__ wdc_W, // [V][HIDDEN]
                                 const float* __restrict__ wdc_b, // [V]
                                 const int* __restrict__ dec_len, // [B]
                                 int t,
                                 float* __restrict__ pred)        // [B][T][V]
{
    __shared__ float htile[16 * HIDDEN];   // 32 KB

    const int lane  = threadIdx.x & 31;
    const int wave  = threadIdx.x >> 5;
    const int btile = blockIdx.x;                 // 0..3
    const int vtile = blockIdx.y * 8 + wave;      // 0..1999
    const int half  = lane >> 4;
    const int l15   = lane & 15;
    const int bcol  = vtile * 16 + l15;           // vocab col

#if HAVE_TDM
    if (threadIdx.x == 0) {
        tdm_load_tile_16rows(h + (size_t)btile * 16 * HIDDEN,
                             (uint32_t)(uintptr_t)htile, HIDDEN);
    }
#else
    for (int i = threadIdx.x; i < 16 * HIDDEN; i += blockDim.x)
        htile[i] = h[(size_t)btile * 16 * HIDDEN + i];
#endif
    __syncthreads();
    lds_written_fence(htile);   // TDM wrote LDS; keep the ds loads alive

    v8f acc = {};
    const float* hlds = htile + l15 * HIDDEN;
    const float* wrow = wdc_W + (size_t)bcol * HIDDEN;
    #pragma unroll 8
    for (int k = 0; k < HIDDEN; k += 4) {
        const int ka = k + half * 2;
        if ((k & 127) == 0) {
            __builtin_prefetch(wrow + k + 256, 0, 1);
        }
        v2f a = *(const v2f*)(hlds + ka);   // ds_load_b64 from staged tile
        v2f b = *(const v2f*)(wrow + ka);
        acc = __builtin_amdgcn_wmma_f32_16x16x4_f32(false, a, false, b,
                                                    (short)0, acc, false, false);
    }

    const float bias = wdc_b[bcol];
    #pragma unroll
    for (int r = 0; r < 8; ++r) {
        const int m = btile * 16 + r + half * 8;
        float v = acc[r] + bias;
        if (t >= dec_len[m]) v = 0.0f;   // mask_t
        pred[(size_t)m * TSTEPS * VOCAB + (size_t)t * VOCAB + bcol] = v;
    }
}

// ---------------------------------------------------------------------------
// Host-side orchestration
// ---------------------------------------------------------------------------
extern "C" void kernel_launch(void* const* d_in, const int* in_sizes, int n_in,
                              void* d_out, int out_size, void* d_ws, size_t ws_size,
                              hipStream_t stream) {
    const float* l_total     = (const float*)d_in[0];
    const int*   enc_caps    = (const int*)  d_in[1];
    const int*   cap_len     = (const int*)  d_in[2];
    const float* emb_weight  = (const float*)d_in[3];
    const float* W_ih        = (const float*)d_in[4];
    const float* W_hh        = (const float*)d_in[5];
    const float* b_ih        = (const float*)d_in[6];
    const float* b_hh        = (const float*)d_in[7];
    const float* wdc_W       = (const float*)d_in[8];
    const float* wdc_b       = (const float*)d_in[9];

    float* ws_f = (float*)d_ws;
    int*   ws_i = (int*)(ws_f + WS_FLOATS);

    float* lt_sorted = ws_f + WS_LT;
    float* h         = ws_f + WS_H;
    float* c         = ws_f + WS_C;
    float* gates     = ws_f + WS_GATES;
    float* x         = ws_f + WS_X;
    int* sort_ind    = ws_i + WSI_SORT;
    int* dec_len     = ws_i + WSI_DLEN;
    int* caps_sorted = ws_i + WSI_CAPS;

    float* out      = (float*)d_out;
    float* out_tail = out + OUT_PRED_ELEMS;

    prep_sort_kernel<<<1, BATCH, 0, stream>>>(enc_caps, cap_len,
                                              sort_ind, dec_len, caps_sorted,
                                              out_tail);
    prep_lt_kernel<<<BATCH, 256, 0, stream>>>(l_total, sort_ind, lt_sorted, h, c);

    for (int t = 0; t < TSTEPS; ++t) {
        build_x_kernel<<<BATCH, 256, 0, stream>>>(emb_weight, caps_sorted,
                                                  lt_sorted, t, x);
        gates_wmma_kernel<<<dim3(4, GATES4 / (16 * 8)), 256, 0, stream>>>(
            x, h, W_ih, W_hh, b_ih, b_hh, gates);
        lstm_update_kernel<<<(BATCH * HIDDEN + 255) / 256, 256, 0, stream>>>(
            gates, h, c);
        pred_wmma_kernel<<<dim3(4, VOCAB / (16 * 8)), 256, 0, stream>>>(
            h, wdc_W, wdc_b, dec_len, t, out);
    }
}